// NeuralODE_32435593019540
// MI455X (gfx1250) — compile-verified
//
#include <hip/hip_runtime.h>
#include <math.h>

typedef __attribute__((ext_vector_type(2))) float v2f;
typedef __attribute__((ext_vector_type(8))) float v8f;

#define HDIM 128
#define WSTR 132   // padded k-stride for transposed weight rows
#define HSTR 132   // padded col-stride for activation buffer rows
#define ASTR 20    // padded stride for a_nn output buffer

// ---- LDS layout (in floats) ----
#define OFF_W0T   0                              // [128][8]   (K padded 6->8)
#define OFF_W1T   (OFF_W0T + HDIM*8)             // [128][WSTR]
#define OFF_W2T   (OFF_W1T + HDIM*WSTR)          // [16][WSTR] (N padded 3->16)
#define OFF_WAVE  (OFF_W2T + 16*WSTR)
#define WAVE_SN   0                               // [32][8]
#define WAVE_HB   (WAVE_SN + 32*8)                // [32][HSTR]
#define WAVE_AN   (WAVE_HB + 32*HSTR)             // [32][ASTR]
#define WAVE_FLOATS (WAVE_AN + 32*ASTR)
#define NWAVES 2
#define LDS_FLOATS (OFF_WAVE + NWAVES*WAVE_FLOATS)

// ---- physics constants (double-folded, matching JAX constant folding) ----
constexpr double dMU = 398600.4418;
constexpr double dRE = 6378.137;
constexpr double dJ2 = 0.00108262668;
constexpr double dJ3 = -2.53265649e-06;
constexpr double dJ4 = -1.61962159e-06;
constexpr double dJ5 = -2.27296083e-07;
constexpr float KMU  = (float)dMU;
constexpr float C2K  = (float)(-1.5  * dJ2 * dMU * dRE*dRE);
constexpr float C3K  = (float)(-2.5  * dJ3 * dMU * dRE*dRE*dRE);
constexpr float C4K  = (float)(1.875 * dJ4 * dMU * dRE*dRE*dRE*dRE);
constexpr float C5K  = (float)(0.375 * dJ5 * dMU * dRE*dRE*dRE*dRE*dRE);
constexpr float C5B  = (float)(1.875 * dJ5 * dMU * dRE*dRE*dRE*dRE*dRE);
constexpr float F703 = (float)(70.0/3.0);
constexpr float R_REF_F = 7000.0f;
constexpr float V_REF_F = 7.5f;
constexpr float A_REF_F = (float)(7.5*7.5/7000.0);
constexpr float DT_F = 60.0f;

__device__ __forceinline__ v8f wmma4(v2f a, v2f b, v8f c) {
  // D(16x16,f32) = A(16x4,f32) x B(4x16,f32) + C   -> v_wmma_f32_16x16x4_f32
  return __builtin_amdgcn_wmma_f32_16x16x4_f32(
      false, a, false, b, (short)0, c, false, false);
}

// Branchless tanh. Prefer the gfx1250 hardware trans op v_tanh_f32; fall back
// to XLA's rational minimax approximation (what the JAX reference computes),
// implemented with clamps + select only (v_cndmask, no EXEC branches).
__device__ __forceinline__ float fast_tanh(float x) {
#if __has_builtin(__builtin_amdgcn_tanhf)
  return __builtin_amdgcn_tanhf(x);
#else
  const float kMax = 7.99881172180175781f;
  float xc = fminf(fmaxf(x, -kMax), kMax);
  float x2 = xc * xc;
  float p = fmaf(x2, -2.76076847742355e-16f, 2.00018790482477e-13f);
  p = fmaf(x2, p, -8.60467152213735e-11f);
  p = fmaf(x2, p, 5.12229709037114e-08f);
  p = fmaf(x2, p, 1.48572235717979e-05f);
  p = fmaf(x2, p, 6.37261928875436e-04f);
  p = fmaf(x2, p, 4.89352455891786e-03f);
  p = xc * p;
  float q = fmaf(x2, 1.19825839466702e-06f, 1.18534705686654e-04f);
  q = fmaf(x2, q, 2.26843463243900e-03f);
  q = fmaf(x2, q, 4.89352518554385e-03f);
  float r = p / q;
  return (fabsf(x) < 0.0004f) ? x : r;
#endif
}

__global__ __launch_bounds__(64) void neural_ode_kernel(
    const float* __restrict__ states0,
    const float* __restrict__ W0, const float* __restrict__ b0,
    const float* __restrict__ W1, const float* __restrict__ b1,
    const float* __restrict__ W2, const float* __restrict__ b2,
    const float* __restrict__ log_scale,
    const int* __restrict__ n_eval_p, const int* __restrict__ spe_p,
    float* __restrict__ out, int B)
{
  extern __shared__ float lds[];
  const int tid = threadIdx.x;
  const int nth = blockDim.x;

  // ---- load weights (transposed: output-col major, K contiguous) ----
  for (int i = tid; i < HDIM*8; i += nth) {
    int n = i >> 3, k = i & 7;
    lds[OFF_W0T + n*8 + k] = (k < 6) ? W0[k*HDIM + n] : 0.0f;
  }
  for (int i = tid; i < HDIM*HDIM; i += nth) {
    int k = i >> 7, n = i & 127;
    lds[OFF_W1T + n*WSTR + k] = W1[i];
  }
  for (int i = tid; i < 16*HDIM; i += nth) {
    int n = i >> 7, k = i & 127;
    lds[OFF_W2T + n*WSTR + k] = (n < 3) ? W2[k*3 + n] : 0.0f;
  }
  __syncthreads();

  const int wave = tid >> 5;
  const int lane = tid & 31;
  const int m    = lane & 15;
  const int hi   = lane >> 4;
  const int h2   = hi << 1;

  float* snb = &lds[OFF_WAVE + wave*WAVE_FLOATS + WAVE_SN];
  float* hb  = &lds[OFF_WAVE + wave*WAVE_FLOATS + WAVE_HB];
  float* anb = &lds[OFF_WAVE + wave*WAVE_FLOATS + WAVE_AN];
  const float* w0t = &lds[OFF_W0T];
  const float* w1t = &lds[OFF_W1T];
  const float* w2t = &lds[OFF_W2T];

  // per-lane bias registers (column = 16*t + m in the D layout)
  float b0r[8], b1r[8];
  #pragma unroll
  for (int t = 0; t < 8; ++t) { b0r[t] = b0[t*16 + m]; b1r[t] = b1[t*16 + m]; }
  const float b2v  = (m < 3) ? b2[m] : 0.0f;
  const float scale = expf(log_scale[0]) * A_REF_F;

  const int spe   = spe_p[0];
  const int nev   = n_eval_p[0];
  const int total = spe * nev;

  const int sat  = blockIdx.x * (NWAVES*32) + wave*32 + lane;
  const int satc = (sat < B) ? sat : (B - 1);
  float px = states0[satc*6+0], py = states0[satc*6+1], pz = states0[satc*6+2];
  float vx = states0[satc*6+3], vy = states0[satc*6+4], vz = states0[satc*6+5];

  const v8f vzero = {0,0,0,0,0,0,0,0};

  // dynamics: acceleration = J2..J5 gravity + MLP correction
  auto dyn_acc = [&](float sx, float sy, float sz,
                     float svx, float svy, float svz,
                     float& ax, float& ay, float& az) {
    // ---- gravity (per-lane f32 VALU) ----
    float r2 = sx*sx + sy*sy + sz*sz;
    float r  = sqrtf(r2);
    float r3 = r2*r, r5 = r2*r3, r7 = r2*r5, r9 = r2*r7;
    float z2 = sz*sz;
    float zr2 = z2 / r2;
    float z4r4 = zr2*zr2;
    float gmr3 = -KMU / r3;
    float gx = gmr3*sx, gy = gmr3*sy, gz = gmr3*sz;
    float c2 = C2K / r5;
    float f2 = 1.0f - 5.0f*zr2;
    gx += c2*sx*f2; gy += c2*sy*f2; gz += c2*sz*(3.0f - 5.0f*zr2);
    float c3 = C3K / r7;
    float t3 = 3.0f*sz - 7.0f*sz*zr2;
    gx += c3*sx*t3; gy += c3*sy*t3;
    gz += c3*(6.0f*z2 - 7.0f*z2*zr2 - 0.6f*r2);
    float c4 = C4K / r7;
    float t4 = 1.0f - 14.0f*zr2 + 21.0f*z4r4;
    gx += c4*sx*t4; gy += c4*sy*t4;
    gz += c4*sz*(5.0f - F703*zr2 + 21.0f*z4r4);
    float c5 = C5K / r9;
    float t5 = 35.0f - 210.0f*zr2 + 231.0f*z4r4;
    gx += c5*sx*sz*t5; gy += c5*sy*sz*t5;
    gz += c5*z2*(105.0f - 315.0f*zr2 + 231.0f*z4r4) - C5B / r7;

    // ---- MLP input: sn = (pos/R_REF, vel/V_REF), K padded to 8 ----
    snb[lane*8+0] = sx * (1.0f/R_REF_F);
    snb[lane*8+1] = sy * (1.0f/R_REF_F);
    snb[lane*8+2] = sz * (1.0f/R_REF_F);
    snb[lane*8+3] = svx * (1.0f/V_REF_F);
    snb[lane*8+4] = svy * (1.0f/V_REF_F);
    snb[lane*8+5] = svz * (1.0f/V_REF_F);
    snb[lane*8+6] = 0.0f;
    snb[lane*8+7] = 0.0f;
    __syncthreads();

    v8f acc0[8], acc1[8];
    #pragma unroll
    for (int t = 0; t < 8; ++t) { acc0[t] = vzero; acc1[t] = vzero; }

    // ---- layer 1: (16x8) x (8x128), two M-tiles ----
    #pragma unroll
    for (int kk = 0; kk < 2; ++kk) {
      const int col = 4*kk + h2;
      v2f a0 = *(const v2f*)&snb[m*8 + col];
      v2f a1 = *(const v2f*)&snb[(16+m)*8 + col];
      #pragma unroll
      for (int t = 0; t < 8; ++t) {
        v2f bt = *(const v2f*)&w0t[(t*16+m)*8 + col];
        acc0[t] = wmma4(a0, bt, acc0[t]);
        acc1[t] = wmma4(a1, bt, acc1[t]);
      }
    }
    __syncthreads();
    #pragma unroll
    for (int t = 0; t < 8; ++t) {
      #pragma unroll
      for (int j = 0; j < 8; ++j) {
        hb[(j + 8*hi)*HSTR + t*16 + m]      = fast_tanh(acc0[t][j] + b0r[t]);
        hb[(16 + j + 8*hi)*HSTR + t*16 + m] = fast_tanh(acc1[t][j] + b0r[t]);
      }
    }
    __syncthreads();

    // ---- layer 2: (16x128) x (128x128), two M-tiles share B tiles ----
    #pragma unroll
    for (int t = 0; t < 8; ++t) { acc0[t] = vzero; acc1[t] = vzero; }
    #pragma unroll 4
    for (int kk = 0; kk < 32; ++kk) {
      const int col = 4*kk + h2;
      v2f a0 = *(const v2f*)&hb[m*HSTR + col];
      v2f a1 = *(const v2f*)&hb[(16+m)*HSTR + col];
      #pragma unroll
      for (int t = 0; t < 8; ++t) {
        v2f bt = *(const v2f*)&w1t[(t*16+m)*WSTR + col];
        acc0[t] = wmma4(a0, bt, acc0[t]);
        acc1[t] = wmma4(a1, bt, acc1[t]);
      }
    }
    __syncthreads();
    #pragma unroll
    for (int t = 0; t < 8; ++t) {
      #pragma unroll
      for (int j = 0; j < 8; ++j) {
        hb[(j + 8*hi)*HSTR + t*16 + m]      = fast_tanh(acc0[t][j] + b1r[t]);
        hb[(16 + j + 8*hi)*HSTR + t*16 + m] = fast_tanh(acc1[t][j] + b1r[t]);
      }
    }
    __syncthreads();

    // ---- layer 3: (16x128) x (128x16) [cols 3..15 are zero pad] ----
    v8f c0 = vzero, c1 = vzero;
    #pragma unroll 4
    for (int kk = 0; kk < 32; ++kk) {
      const int col = 4*kk + h2;
      v2f a0 = *(const v2f*)&hb[m*HSTR + col];
      v2f a1 = *(const v2f*)&hb[(16+m)*HSTR + col];
      v2f bt = *(const v2f*)&w2t[m*WSTR + col];
      c0 = wmma4(a0, bt, c0);
      c1 = wmma4(a1, bt, c1);
    }
    __syncthreads();
    #pragma unroll
    for (int j = 0; j < 8; ++j) {
      anb[(j + 8*hi)*ASTR + m]      = c0[j] + b2v;
      anb[(16 + j + 8*hi)*ASTR + m] = c1[j] + b2v;
    }
    __syncthreads();
    float anx = anb[lane*ASTR + 0] * scale;
    float any = anb[lane*ASTR + 1] * scale;
    float anz = anb[lane*ASTR + 2] * scale;

    ax = gx + anx; ay = gy + any; az = gz + anz;
  };

  // ---- RK4 time loop (uniform across all waves: EXEC stays full) ----
  const float hdt = 0.5f * DT_F;
  const float s6  = DT_F / 6.0f;
  for (int step = 0; step < total; ++step) {
    float a1x,a1y,a1z, a2x,a2y,a2z, a3x,a3y,a3z, a4x,a4y,a4z;
    dyn_acc(px,py,pz, vx,vy,vz, a1x,a1y,a1z);
    float p2x=px+hdt*vx,  p2y=py+hdt*vy,  p2z=pz+hdt*vz;
    float v2x=vx+hdt*a1x, v2y=vy+hdt*a1y, v2z=vz+hdt*a1z;
    dyn_acc(p2x,p2y,p2z, v2x,v2y,v2z, a2x,a2y,a2z);
    float p3x=px+hdt*v2x, p3y=py+hdt*v2y, p3z=pz+hdt*v2z;
    float v3x=vx+hdt*a2x, v3y=vy+hdt*a2y, v3z=vz+hdt*a2z;
    dyn_acc(p3x,p3y,p3z, v3x,v3y,v3z, a3x,a3y,a3z);
    float p4x=px+DT_F*v3x, p4y=py+DT_F*v3y, p4z=pz+DT_F*v3z;
    float v4x=vx+DT_F*a3x, v4y=vy+DT_F*a3y, v4z=vz+DT_F*a3z;
    dyn_acc(p4x,p4y,p4z, v4x,v4y,v4z, a4x,a4y,a4z);

    px += s6*(vx + 2.0f*v2x + 2.0f*v3x + v4x);
    py += s6*(vy + 2.0f*v2y + 2.0f*v3y + v4y);
    pz += s6*(vz + 2.0f*v2z + 2.0f*v3z + v4z);
    vx += s6*(a1x + 2.0f*a2x + 2.0f*a3x + a4x);
    vy += s6*(a1y + 2.0f*a2y + 2.0f*a3y + a4y);
    vz += s6*(a1z + 2.0f*a2z + 2.0f*a3z + a4z);

    if (((step + 1) % spe) == 0 && sat < B) {
      int e = (step + 1) / spe - 1;
      int base = (sat*nev + e) * 6;
      out[base+0] = px; out[base+1] = py; out[base+2] = pz;
      out[base+3] = vx; out[base+4] = vy; out[base+5] = vz;
    }
  }
}

extern "C" void kernel_launch(void* const* d_in, const int* in_sizes, int n_in,
                              void* d_out, int out_size, void* d_ws, size_t ws_size,
                              hipStream_t stream) {
  const float* states0   = (const float*)d_in[0];
  const float* W0        = (const float*)d_in[1];
  const float* b0        = (const float*)d_in[2];
  const float* W1        = (const float*)d_in[3];
  const float* b1        = (const float*)d_in[4];
  const float* W2        = (const float*)d_in[5];
  const float* b2        = (const float*)d_in[6];
  const float* log_scale = (const float*)d_in[7];
  const int*   n_eval    = (const int*)d_in[8];
  const int*   spe       = (const int*)d_in[9];
  float* out = (float*)d_out;

  const int B = in_sizes[0] / 6;
  const int satsPerWG = NWAVES * 32;
  const int grid = (B + satsPerWG - 1) / satsPerWG;
  const size_t shmem = (size_t)LDS_FLOATS * sizeof(float);

  neural_ode_kernel<<<grid, NWAVES*32, shmem, stream>>>(
      states0, W0, b0, W1, b1, W2, b2, log_scale, n_eval, spe, out, B);
}